// Conv_82506321756833
// MI455X (gfx1250) — compile-verified
//
#include <hip/hip_runtime.h>
#include <hip/hip_bf16.h>
#include <math.h>

#define HDIM 128
#define BN_EPS 1e-5f

typedef float v2f __attribute__((ext_vector_type(2)));
typedef float v8f __attribute__((ext_vector_type(8)));

__device__ __forceinline__ float gelu_erf(float x) {
    // torch nn.GELU default: 0.5*x*(1+erf(x/sqrt(2)))
    return 0.5f * x * (1.0f + erff(x * 0.70710678118654752f));
}

// out[row, n] = [residual +] GELU( [BN]( X @ W + bias ) )
// One block = 256 threads = 8 waves; block covers a 16-row strip, wave w owns
// the 16-column tile starting at 16*w. K = HDIM = 128 -> 32 WMMA f32 16x16x4
// steps, split over two accumulators to break the D->C dependency chain.
__global__ void wmma_gemm_fused(const float* __restrict__ X,
                                const float* __restrict__ W,
                                const float* __restrict__ bias,
                                const float* __restrict__ g,
                                const float* __restrict__ be,
                                const float* __restrict__ mean,
                                const float* __restrict__ var,
                                const float* __restrict__ residual,
                                float* __restrict__ out,
                                int M, int doBN, int doResid)
{
    const int lane  = threadIdx.x & 31;
    const int wave  = threadIdx.x >> 5;       // 0..7
    const int tileM = blockIdx.x * 16;
    const int tileN = wave * 16;
    const int l15   = lane & 15;              // M index (A) / N index (B,C,D)
    const int khalf = (lane >> 4) << 1;       // 0 (lanes 0-15) or 2 (lanes 16-31)

    int rowA = tileM + l15;
    if (rowA >= M) rowA = M - 1;              // clamp: keep EXEC all-ones for WMMA
    const float* __restrict__ arow = X + (size_t)rowA * HDIM;
    const float* __restrict__ bcol = W + tileN + l15;

    v8f c0 = {}, c1 = {};
    #pragma unroll 4
    for (int k = 0; k < HDIM / 8; ++k) {
        // even k-step -> c0
        {
            const int kb = (2 * k) * 4 + khalf;
            v2f a, b;
            a.x = arow[kb];
            a.y = arow[kb + 1];
            b.x = bcol[(size_t)kb * HDIM];
            b.y = bcol[(size_t)(kb + 1) * HDIM];
            c0 = __builtin_amdgcn_wmma_f32_16x16x4_f32(false, a, false, b,
                                                       (short)0, c0, false, false);
        }
        // odd k-step -> c1
        {
            const int kb = (2 * k + 1) * 4 + khalf;
            v2f a, b;
            a.x = arow[kb];
            a.y = arow[kb + 1];
            b.x = bcol[(size_t)kb * HDIM];
            b.y = bcol[(size_t)(kb + 1) * HDIM];
            c1 = __builtin_amdgcn_wmma_f32_16x16x4_f32(false, a, false, b,
                                                       (short)0, c1, false, false);
        }
    }

    const int n  = tileN + l15;               // this lane's output column
    const float bv = bias[n];
    float scale = 1.0f, shift = 0.0f;
    if (doBN) {
        scale = g[n] * rsqrtf(var[n] + BN_EPS);
        shift = be[n] - mean[n] * scale;
    }
    const int rbase = tileM + ((lane >> 4) << 3);   // +8 rows for upper half-wave
    #pragma unroll
    for (int i = 0; i < 8; ++i) {
        const int row = rbase + i;
        if (row >= M) continue;
        float val = c0[i] + c1[i] + bv;
        if (doBN) val = val * scale + shift;
        val = gelu_erf(val);
        if (doResid) val += residual[(size_t)row * HDIM + n];
        out[(size_t)row * HDIM + n] = val;
    }
}

// One wave per edge: msg = xp[src[e]] * bases[e]; atomic scatter-add into accum[dst[e]].
// Each lane handles 4 contiguous floats (float4 loads + 4 global_atomic_add_f32).
__global__ void edge_scatter(const float* __restrict__ xp,
                             const float* __restrict__ bases,
                             const int* __restrict__ src,
                             const int* __restrict__ dst,
                             float* __restrict__ accum,
                             int E)
{
    const int lane = threadIdx.x & 31;
    const int e = blockIdx.x * 8 + (threadIdx.x >> 5);
    if (e >= E) return;
    const int s = src[e];
    const int d = dst[e];
    const float4 xv = *(const float4*)(xp + (size_t)s * HDIM + lane * 4);
    const float4 bv = *(const float4*)(bases + (size_t)e * HDIM + lane * 4);
    float* outp = accum + (size_t)d * HDIM + lane * 4;
    unsafeAtomicAdd(outp + 0, xv.x * bv.x);
    unsafeAtomicAdd(outp + 1, xv.y * bv.y);
    unsafeAtomicAdd(outp + 2, xv.z * bv.z);
    unsafeAtomicAdd(outp + 3, xv.w * bv.w);
}

extern "C" void kernel_launch(void* const* d_in, const int* in_sizes, int n_in,
                              void* d_out, int out_size, void* d_ws, size_t ws_size,
                              hipStream_t stream)
{
    const float* x_feat = (const float*)d_in[0];
    const float* bases  = (const float*)d_in[1];
    const int*   src    = (const int*)d_in[2];
    const int*   dst    = (const int*)d_in[3];
    const float* W1 = (const float*)d_in[4];
    const float* b1 = (const float*)d_in[5];
    const float* W2 = (const float*)d_in[6];
    const float* b2 = (const float*)d_in[7];
    const float* W3 = (const float*)d_in[8];
    const float* b3 = (const float*)d_in[9];
    const float* g1 = (const float*)d_in[10];
    const float* be1= (const float*)d_in[11];
    const float* m1 = (const float*)d_in[12];
    const float* v1 = (const float*)d_in[13];
    const float* g2 = (const float*)d_in[14];
    const float* be2= (const float*)d_in[15];
    const float* m2 = (const float*)d_in[16];
    const float* v2 = (const float*)d_in[17];

    const int M = in_sizes[0] / HDIM;   // 50000 nodes
    const int E = in_sizes[2];          // 800000 edges

    float* xp   = (float*)d_ws;                       // [M, H]
    float* x_ws = xp   + (size_t)M * HDIM;            // [M, H]
    float* y_ws = x_ws + (size_t)M * HDIM;            // [M, H]
    float* outp = (float*)d_out;

    const int mtiles = (M + 15) / 16;
    dim3 blk(256);

    // xp = GELU(x_feat @ W1 + b1)
    wmma_gemm_fused<<<mtiles, blk, 0, stream>>>(x_feat, W1, b1,
        nullptr, nullptr, nullptr, nullptr, nullptr, xp, M, 0, 0);

    // x = x_feat, then x += segment_sum(xp[src] * bases -> dst)
    hipMemcpyAsync(x_ws, x_feat, (size_t)M * HDIM * sizeof(float),
                   hipMemcpyDeviceToDevice, stream);
    edge_scatter<<<(E + 7) / 8, blk, 0, stream>>>(xp, bases, src, dst, x_ws, E);

    // y = GELU(BN1(x @ W2 + b2))
    wmma_gemm_fused<<<mtiles, blk, 0, stream>>>(x_ws, W2, b2,
        g1, be1, m1, v1, nullptr, y_ws, M, 1, 0);

    // out = x + GELU(BN2(y @ W3 + b3))
    wmma_gemm_fused<<<mtiles, blk, 0, stream>>>(y_ws, W3, b3,
        g2, be2, m2, v2, x_ws, outp, M, 1, 1);
}